// RESUS_RR_89781996355696
// MI455X (gfx1250) — compile-verified
//
#include <hip/hip_runtime.h>
#include <hip/hip_bf16.h>
#include <math.h>

#define BDIM 256
#define KSUP 30
#define NF 23
#define EDIM 16
#define HD 256
#define NROW 31            // K+1
#define FED 368            // NF*EDIM
#define EH_STRIDE 369      // odd -> conflict-free A-fragment loads
#define G_STRIDE 257       // odd -> conflict-free
#define S_STRIDE 33
#define A_STRIDE 33

typedef __attribute__((ext_vector_type(2))) float v2f;
typedef __attribute__((ext_vector_type(8))) float v8f;

// Repack enc_W (368x256, row-major) into K-groups of 4:
// packed[((k>>2)*HD + col)*4 + (k&3)] = enc_W[k*HD + col]
// => a WMMA B-fragment's (kr, kr+1) pair is one contiguous 8B load.
__global__ __launch_bounds__(256)
void pack_encW_kernel(const float* __restrict__ encW, float* __restrict__ packed)
{
    int idx = blockIdx.x * 256 + threadIdx.x;
    if (idx >= FED * HD) return;
    int k = idx / HD, col = idx - k * HD;
    packed[(((k >> 2) * HD + col) << 2) + (k & 3)] = encW[idx];
}

template <bool PACKED>
__global__ __launch_bounds__(BDIM)
void resus_fused_kernel(const int*   __restrict__ feature_ids,
                        const float* __restrict__ feature_vals,
                        const int*   __restrict__ x_id_support,
                        const float* __restrict__ x_val_support,
                        const float* __restrict__ y_support,
                        const float* __restrict__ pred_emb,
                        const float* __restrict__ pred_w,
                        const float* __restrict__ pred_b,
                        const float* __restrict__ enc_emb,
                        const float* __restrict__ encw,      // packed or raw per PACKED
                        const float* __restrict__ lambda_l,
                        const float* __restrict__ adj_scale,
                        const float* __restrict__ adj_bias,
                        float* __restrict__ out)
{
    __shared__ int   idsLds[NROW * NF];
    __shared__ float valsLds[NROW * NF];
    __shared__ float yLds[KSUP];
    __shared__ float maskLds[KSUP];
    __shared__ float ehLds[32 * EH_STRIDE];   // 47.2 KB, row 31 zero-padded
    __shared__ float gLds[32 * G_STRIDE];     // 32.9 KB, row 0 = g_hat, 31 = 0
    __shared__ float sLds[32 * S_STRIDE];     // S = G G^T (4.2 KB)
    __shared__ float logitsPart[NROW * 8];
    __shared__ float logitsLds[NROW];
    __shared__ float Am[KSUP * A_STRIDE];     // augmented [gram | y_res]
    __shared__ float alphaLds[KSUP];

    const int b   = blockIdx.x;
    const int tid = threadIdx.x;

    // ---- Phase 0: stage ids / vals / y for this batch element ----
    for (int idx = tid; idx < NROW * NF; idx += BDIM) {
        int r = idx / NF, f = idx - r * NF;
        int id; float val;
        if (r == 0) { id = feature_ids[b * NF + f]; val = feature_vals[b * NF + f]; }
        else {
            int o = (b * KSUP + (r - 1)) * NF + f;
            id = x_id_support[o]; val = x_val_support[o];
        }
        idsLds[idx] = id; valsLds[idx] = val;
    }
    if (tid < KSUP) {
        float y = y_support[b * KSUP + tid];
        yLds[tid]    = y;
        maskLds[tid] = (y != -1.0f) ? 1.0f : 0.0f;
    }
    __syncthreads();

    // ---- Phase 1: build eh (32x368 fp32 in LDS) + logits partial dots ----
    for (int idx = tid; idx < 32 * FED; idx += BDIM) {
        int r = idx / FED, j = idx - r * FED;
        float v = 0.0f;
        if (r < NROW) {
            int f = j >> 4, e = j & 15;
            v = enc_emb[idsLds[r * NF + f] * EDIM + e] * valsLds[r * NF + f];
        }
        ehLds[r * EH_STRIDE + j] = v;
    }
    if (tid < NROW * 8) {                       // 8 partial lanes per row
        int r = tid >> 3, part = tid & 7;
        float s = 0.0f;
        for (int j = part * 46; j < part * 46 + 46; ++j) {   // 8*46 == 368
            int f = j >> 4, e = j & 15;
            float ph = pred_emb[idsLds[r * NF + f] * EDIM + e] * valsLds[r * NF + f];
            s += ph * pred_w[j];
        }
        logitsPart[tid] = s;
    }
    __syncthreads();

    if (tid < NROW) {
        float s = pred_b[0];
        for (int p = 0; p < 8; ++p) s += logitsPart[tid * 8 + p];
        logitsLds[tid] = s;
    }

    // ---- Phase 2: WMMA f32 GEMM  g(32x256) = ehPad(32x368) @ enc_W(368x256)
    // 8 waves: wave&1 -> M-tile, (wave>>1)*4 -> 4 N-tiles. EXEC all-1s here.
    {
        const int wave   = tid >> 5;
        const int lane   = tid & 31;
        const int mt     = wave & 1;
        const int ntBase = (wave >> 1) << 2;
        const int hs     = lane >> 4;           // half-wave select
        const int l16    = lane & 15;
        const int aRow   = mt * 16 + l16;

        v8f acc0 = {}, acc1 = {}, acc2 = {}, acc3 = {};
        for (int k0 = 0; k0 < FED; k0 += 4) {   // 92 k-steps of K=4
            const int kr = k0 + 2 * hs;
            v2f a;
            a.x = ehLds[aRow * EH_STRIDE + kr + 0];
            a.y = ehLds[aRow * EH_STRIDE + kr + 1];
            v2f b0, b1, b2, b3;
            if (PACKED) {
                // packed float2 index: ((k0>>2)*HD + col)*2 + hs
                const float2* wp = (const float2*)encw;
                const int base = ((k0 >> 2) * HD + ntBase * 16 + l16) * 2 + hs;
                float2 t0 = wp[base];      b0.x = t0.x; b0.y = t0.y;
                float2 t1 = wp[base + 32]; b1.x = t1.x; b1.y = t1.y;  // +16 cols
                float2 t2 = wp[base + 64]; b2.x = t2.x; b2.y = t2.y;
                float2 t3 = wp[base + 96]; b3.x = t3.x; b3.y = t3.y;
            } else {
                const float* bp = encw + (size_t)kr * HD + ntBase * 16 + l16;
                b0.x = bp[0];  b0.y = bp[HD];
                b1.x = bp[16]; b1.y = bp[HD + 16];
                b2.x = bp[32]; b2.y = bp[HD + 32];
                b3.x = bp[48]; b3.y = bp[HD + 48];
            }
            acc0 = __builtin_amdgcn_wmma_f32_16x16x4_f32(false, a, false, b0, (short)0, acc0, false, false);
            acc1 = __builtin_amdgcn_wmma_f32_16x16x4_f32(false, a, false, b1, (short)0, acc1, false, false);
            acc2 = __builtin_amdgcn_wmma_f32_16x16x4_f32(false, a, false, b2, (short)0, acc2, false, false);
            acc3 = __builtin_amdgcn_wmma_f32_16x16x4_f32(false, a, false, b3, (short)0, acc3, false, false);
        }
        const int rBase = mt * 16 + 8 * hs;     // C/D layout: VGPR i -> M=i (+8 hi half)
        #pragma unroll
        for (int i = 0; i < 8; ++i) {
            gLds[(rBase + i) * G_STRIDE + ntBase * 16      + l16] = acc0[i];
            gLds[(rBase + i) * G_STRIDE + ntBase * 16 + 16 + l16] = acc1[i];
            gLds[(rBase + i) * G_STRIDE + ntBase * 16 + 32 + l16] = acc2[i];
            gLds[(rBase + i) * G_STRIDE + ntBase * 16 + 48 + l16] = acc3[i];
        }
    }
    __syncthreads();

    // ---- Phase 3a: S(32x32) = G(32x256) @ G^T via WMMA, 4 waves x 1 tile ----
    // B-fragment of G·G^T is G re-read with the N-tile's row index: same
    // addressing as A (G_STRIDE odd -> conflict-free).
    {
        const int wave = tid >> 5;
        if (wave < 4) {
            const int lane = tid & 31;
            const int mt = wave & 1, nt = wave >> 1;
            const int hs = lane >> 4, l16 = lane & 15;
            const int aRow = mt * 16 + l16;
            const int bRow = nt * 16 + l16;
            v8f acc = {};
            for (int k0 = 0; k0 < HD; k0 += 4) {   // 64 k-steps
                const int kr = k0 + 2 * hs;
                v2f a, bb;
                a.x  = gLds[aRow * G_STRIDE + kr]; a.y  = gLds[aRow * G_STRIDE + kr + 1];
                bb.x = gLds[bRow * G_STRIDE + kr]; bb.y = gLds[bRow * G_STRIDE + kr + 1];
                acc = __builtin_amdgcn_wmma_f32_16x16x4_f32(false, a, false, bb, (short)0, acc, false, false);
            }
            const int rBase = mt * 16 + 8 * hs;
            #pragma unroll
            for (int i = 0; i < 8; ++i)
                sLds[(rBase + i) * S_STRIDE + nt * 16 + l16] = acc[i];
        }
    }
    __syncthreads();

    // ---- Phase 3b: assemble augmented system from S ----
    if (tid < KSUP) {
        float yr = yLds[tid] - 1.0f / (1.0f + __expf(-logitsLds[tid + 1]));
        Am[tid * A_STRIDE + KSUP] = yr;         // rhs column
    }
    {
        const float lam = fabsf(lambda_l[0]);
        for (int e = tid; e < KSUP * KSUP; e += BDIM) {
            int i = e / KSUP, j = e - i * KSUP;
            float v = maskLds[i] * maskLds[j] * (sLds[(i + 1) * S_STRIDE + (j + 1)] + 1.0f);
            if (i == j) v += lam;
            Am[i * A_STRIDE + j] = v;
        }
    }
    __syncthreads();

    // ---- Phase 4: Gaussian elimination (SPD + lambda*I, no pivoting) ----
    for (int k = 0; k < KSUP; ++k) {
        if (tid > k && tid < KSUP) {
            float f = Am[tid * A_STRIDE + k] / Am[k * A_STRIDE + k];
            for (int j = k; j <= KSUP; ++j)
                Am[tid * A_STRIDE + j] -= f * Am[k * A_STRIDE + j];
        }
        __syncthreads();
    }
    for (int ii = KSUP - 1; ii >= 0; --ii) {
        if (tid == ii) {
            float s = Am[ii * A_STRIDE + KSUP];
            for (int j = ii + 1; j < KSUP; ++j) s -= Am[ii * A_STRIDE + j] * alphaLds[j];
            alphaLds[ii] = s / Am[ii * A_STRIDE + ii];
        }
        __syncthreads();
    }

    // ---- Phase 5: out = sum_k alpha_k * mask_k * (S[0][k+1] + 1) ----
    if (tid == 0) {
        float o = 0.0f; int nv = 0;
        for (int k = 0; k < KSUP; ++k) {
            o  += alphaLds[k] * maskLds[k] * (sLds[0 * S_STRIDE + (k + 1)] + 1.0f);
            nv += (maskLds[k] > 0.5f) ? 1 : 0;
        }
        int idx = nv - 1;
        if (idx < 0) idx += KSUP;                     // jnp wrap semantics
        out[b] = o * fabsf(adj_scale[idx]) + adj_bias[idx] + logitsLds[0];
    }
}

extern "C" void kernel_launch(void* const* d_in, const int* in_sizes, int n_in,
                              void* d_out, int out_size, void* d_ws, size_t ws_size,
                              hipStream_t stream) {
    (void)n_in; (void)out_size;
    const int Bn = in_sizes[0] / NF;   // B = 4096
    const bool packed = (ws_size >= (size_t)FED * HD * sizeof(float)) && d_ws;
    if (packed) {
        pack_encW_kernel<<<(FED * HD + 255) / 256, 256, 0, stream>>>(
            (const float*)d_in[9], (float*)d_ws);
        hipLaunchKernelGGL(resus_fused_kernel<true>, dim3(Bn), dim3(BDIM), 0, stream,
                           (const int*)  d_in[0],  (const float*)d_in[1],
                           (const int*)  d_in[2],  (const float*)d_in[3],
                           (const float*)d_in[4],  (const float*)d_in[5],
                           (const float*)d_in[6],  (const float*)d_in[7],
                           (const float*)d_in[8],  (const float*)d_ws,
                           (const float*)d_in[10], (const float*)d_in[11],
                           (const float*)d_in[12], (float*)d_out);
    } else {
        hipLaunchKernelGGL(resus_fused_kernel<false>, dim3(Bn), dim3(BDIM), 0, stream,
                           (const int*)  d_in[0],  (const float*)d_in[1],
                           (const int*)  d_in[2],  (const float*)d_in[3],
                           (const float*)d_in[4],  (const float*)d_in[5],
                           (const float*)d_in[6],  (const float*)d_in[7],
                           (const float*)d_in[8],  (const float*)d_in[9],
                           (const float*)d_in[10], (const float*)d_in[11],
                           (const float*)d_in[12], (float*)d_out);
    }
}